// GCN_58506044506835
// MI455X (gfx1250) — compile-verified
//
#include <hip/hip_runtime.h>
#include <hip/hip_bf16.h>

typedef __attribute__((ext_vector_type(2))) float v2f;
typedef __attribute__((ext_vector_type(8))) float v8f;

#define NNODES 100000
#define NEDGES 1600000
#define NB     64
#define BN_EPS 1e-5f

// ---------------------------------------------------------------- degree / norm
__global__ void k_deg_init(float* deg) {
    int i = blockIdx.x * blockDim.x + threadIdx.x;
    if (i < NNODES) deg[i] = 1.0f;              // self-loop contribution
}
__global__ void k_deg_edges(const long long* __restrict__ col, float* deg) {
    int e = blockIdx.x * blockDim.x + threadIdx.x;
    if (e < NEDGES) atomicAdd(&deg[(int)col[e]], 1.0f);
}
__global__ void k_rsqrt_inplace(float* deg) {
    int i = blockIdx.x * blockDim.x + threadIdx.x;
    if (i < NNODES) deg[i] = rsqrtf(deg[i]);
}

// ---------------------------------------------------------------- layer-1 GEMM (3 -> 16) via f32 WMMA, K padded 3->4
__global__ void k_gemm3_16_wmma(const float* __restrict__ x,
                                const float* __restrict__ W,   // [3][16] row-major
                                float* __restrict__ xw,        // [N][16]
                                int ntiles) {
    int lane = threadIdx.x & 31;
    int wave = threadIdx.x >> 5;
    int tile = blockIdx.x * (blockDim.x >> 5) + wave;
    if (tile >= ntiles) return;                 // wave-uniform; EXEC all-ones below

    int m    = lane & 15;
    int node = tile * 16 + m;

    float a0, a1, b0, b1;
    if (lane < 16) { a0 = x[node * 3 + 0]; a1 = x[node * 3 + 1]; }
    else           { a0 = x[node * 3 + 2]; a1 = 0.0f; }
    int n = lane & 15;
    if (lane < 16) { b0 = W[0 * 16 + n]; b1 = W[1 * 16 + n]; }
    else           { b0 = W[2 * 16 + n]; b1 = 0.0f; }
    v2f A = {a0, a1};
    v2f Bm = {b0, b1};
    v8f acc = {};
    acc = __builtin_amdgcn_wmma_f32_16x16x4_f32(false, A, false, Bm,
                                                (short)0, acc, false, false);
    int mbase = (lane >> 4) * 8;
    float* drow = xw + (size_t)tile * 16 * 16;
    #pragma unroll
    for (int v = 0; v < 8; ++v)
        drow[(size_t)(mbase + v) * 16 + n] = acc[v];
}

// ---------------------------------------------------------------- generic GEMM via f32 WMMA, fully unrolled
template<int Cin, int Cout>
__global__ void k_gemm_wmma_t(const float* __restrict__ hin,
                              const float* __restrict__ W,     // [Cin][Cout] row-major
                              float* __restrict__ xw,          // [N][Cout]
                              int ntiles) {
    int lane = threadIdx.x & 31;
    int wave = threadIdx.x >> 5;
    int tile = blockIdx.x * (blockDim.x >> 5) + wave;
    if (tile >= ntiles) return;                 // wave-uniform

    int m     = lane & 15;
    int n     = lane & 15;
    int khalf = (lane >> 4) * 2;                // 0 for lanes 0-15, 2 for lanes 16-31
    const float* arow = hin + (size_t)(tile * 16 + m) * Cin;

    // A fragments are shared by all N-tiles: load once.
    v2f Afrag[Cin / 4];
    #pragma unroll
    for (int k0 = 0; k0 < Cin; k0 += 4)
        Afrag[k0 / 4] = *(const v2f*)(arow + k0 + khalf);   // 8B aligned

    #pragma unroll
    for (int nt = 0; nt < (Cout >> 4); ++nt) {
        const float* wcol = W + nt * 16 + n;
        v8f acc = {};
        #pragma unroll
        for (int k0 = 0; k0 < Cin; k0 += 4) {
            int kk = k0 + khalf;
            v2f Bm = { wcol[(size_t)kk * Cout],
                       wcol[(size_t)(kk + 1) * Cout] };
            acc = __builtin_amdgcn_wmma_f32_16x16x4_f32(false, Afrag[k0 / 4], false, Bm,
                                                        (short)0, acc, false, false);
        }
        int mbase = (lane >> 4) * 8;
        float* drow = xw + (size_t)tile * 16 * Cout + nt * 16 + n;
        #pragma unroll
        for (int v = 0; v < 8; ++v)
            drow[(size_t)(mbase + v) * Cout] = acc[v];
    }
}

// ---------------------------------------------------------------- aggregation
// init target with bias + self-loop message (replaces zeroing): flat, coalesced
template<int C>
__global__ void k_agg_init_t(const float* __restrict__ xw, const float* __restrict__ dis,
                             const float* __restrict__ bias, float* __restrict__ agg) {
    size_t tid = (size_t)blockIdx.x * blockDim.x + threadIdx.x;
    if (tid >= (size_t)NNODES * C) return;
    int i = (int)(tid / C);
    int c = (int)(tid % C);
    float w = dis[i] * dis[i];
    agg[tid] = bias[c] + xw[tid] * w;
}

// one thread per (edge, channel): index/dis loads broadcast within the wave,
// xw gather and f32 atomics fully coalesced to single L2 lines.
template<int C>
__global__ void k_agg_edges_t(const long long* __restrict__ row, const long long* __restrict__ col,
                              const float* __restrict__ dis, const float* __restrict__ xw,
                              float* __restrict__ agg) {
    size_t tid = (size_t)blockIdx.x * blockDim.x + threadIdx.x;
    if (tid >= (size_t)NEDGES * C) return;
    int e = (int)(tid / C);
    int c = (int)(tid % C);
    int r = (int)row[e], t = (int)col[e];
    float w = dis[r] * dis[t];
    atomicAdd(&agg[(size_t)t * C + c], xw[(size_t)r * C + c] * w);
}

// ---------------------------------------------------------------- BN statistics and fused BN+ReLU
__global__ void k_zero(float* p, int n) {
    int i = blockIdx.x * blockDim.x + threadIdx.x;
    if (i < n) p[i] = 0.0f;
}

// thread t owns channel t%C, node-group t/C: coalesced loads, register accumulate,
// plain LDS combine (no LDS atomics), 2C global atomics per block.
template<int C>
__global__ void k_stats_t(const float* __restrict__ h, float* __restrict__ stats) {
    const int G = 256 / C;                     // thread groups per channel
    __shared__ float ssum[256];
    __shared__ float ssq[256];
    int t = threadIdx.x;
    int c = t % C, g = t / C;
    int base = blockIdx.x * 256;               // 256 nodes per block
    float s = 0.0f, q = 0.0f;
    for (int i = g; i < 256; i += G) {
        int node = base + i;
        if (node < NNODES) {
            float v = h[(size_t)node * C + c];
            s += v; q += v * v;
        }
    }
    ssum[t] = s; ssq[t] = q;
    __syncthreads();
    if (t < C) {
        float S = 0.0f, Q = 0.0f;
        #pragma unroll
        for (int g2 = 0; g2 < G; ++g2) { S += ssum[g2 * C + t]; Q += ssq[g2 * C + t]; }
        atomicAdd(&stats[t],     S);
        atomicAdd(&stats[C + t], Q);
    }
}

__global__ void k_coeff(const float* __restrict__ stats, const float* __restrict__ g,
                        const float* __restrict__ be, float* __restrict__ coeff, int C) {
    int c = blockIdx.x * blockDim.x + threadIdx.x;
    if (c >= C) return;
    float invN = 1.0f / (float)NNODES;
    float mean = stats[c] * invN;
    float var  = stats[C + c] * invN - mean * mean;   // biased variance (torch BN)
    float s    = g[c] * rsqrtf(var + BN_EPS);
    coeff[c]     = s;
    coeff[C + c] = be[c] - mean * s;
}

__global__ void k_bnrelu(float* __restrict__ h, const float* __restrict__ coeff, int C) {
    int i = blockIdx.x * blockDim.x + threadIdx.x;
    if (i >= NNODES * C) return;
    int c = i % C;
    h[i] = fmaxf(0.0f, h[i] * coeff[c] + coeff[C + c]);
}

// ---------------------------------------------------------------- mean pool per batch + FC
__global__ void k_pool(const float* __restrict__ h, const long long* __restrict__ batch,
                       float* __restrict__ pool, float* __restrict__ cnt) {
    size_t tid = (size_t)blockIdx.x * blockDim.x + threadIdx.x;
    if (tid >= (size_t)NNODES * 64) return;
    int i = (int)(tid >> 6);
    int c = (int)(tid & 63);
    int b = (int)batch[i];
    atomicAdd(&pool[(size_t)b * 64 + c], h[tid]);
    if (c == 0) atomicAdd(&cnt[b], 1.0f);
}

__global__ void k_fc(const float* __restrict__ pool, const float* __restrict__ cnt,
                     const float* __restrict__ fcW, const float* __restrict__ fcb,
                     float* __restrict__ out) {
    int j = blockIdx.x * blockDim.x + threadIdx.x;
    if (j >= NB * 10) return;
    int b = j / 10, o = j % 10;
    float inv = 1.0f / fmaxf(cnt[b], 1.0f);
    const float* pr = pool + (size_t)b * 64;
    float acc = fcb[o];
    for (int c = 0; c < 64; ++c) acc += pr[c] * inv * fcW[c * 10 + o];
    out[j] = acc;
}

// ---------------------------------------------------------------- host orchestration
static inline int nblk(long long n, int t) { return (int)((n + t - 1) / t); }

extern "C" void kernel_launch(void* const* d_in, const int* in_sizes, int n_in,
                              void* d_out, int out_size, void* d_ws, size_t ws_size,
                              hipStream_t stream) {
    const float*     x    = (const float*)d_in[0];
    const long long* ei   = (const long long*)d_in[1];
    const long long* bat  = (const long long*)d_in[2];
    const float *W1 = (const float*)d_in[3],  *b1 = (const float*)d_in[4];
    const float *g1 = (const float*)d_in[5],  *be1 = (const float*)d_in[6];
    const float *W2 = (const float*)d_in[7],  *b2 = (const float*)d_in[8];
    const float *g2 = (const float*)d_in[9],  *be2 = (const float*)d_in[10];
    const float *W3 = (const float*)d_in[11], *b3 = (const float*)d_in[12];
    const float *g3 = (const float*)d_in[13], *be3 = (const float*)d_in[14];
    const float *fcW = (const float*)d_in[15], *fcb = (const float*)d_in[16];
    float* out = (float*)d_out;

    const long long* row = ei;            // edge_index[0]
    const long long* col = ei + NEDGES;   // edge_index[1]

    // workspace layout (floats)
    float* w    = (float*)d_ws;
    float* dis  = w;                                   // [N]
    float* P0   = dis + NNODES;                        // [N*64] agg / h
    float* P1   = P0 + (size_t)NNODES * 64;            // [N*64] xw
    float* stat = P1 + (size_t)NNODES * 64;            // [128]
    float* cf   = stat + 128;                          // [128]
    float* pool = cf + 128;                            // [B*64]
    float* cnt  = pool + NB * 64;                      // [B]

    const int T = 256;
    const int ntiles = NNODES / 16;                    // 6250 exactly
    const int gemmBlocks = nblk(ntiles, T / 32);       // 8 waves per block

    // ---- degrees / symmetric norm
    k_deg_init<<<nblk(NNODES, T), T, 0, stream>>>(dis);
    k_deg_edges<<<nblk(NEDGES, T), T, 0, stream>>>(col, dis);
    k_rsqrt_inplace<<<nblk(NNODES, T), T, 0, stream>>>(dis);

    // ---- layer 1: 3 -> 16
    k_gemm3_16_wmma<<<gemmBlocks, T, 0, stream>>>(x, W1, P1, ntiles);
    k_agg_init_t<16><<<nblk((long long)NNODES * 16, T), T, 0, stream>>>(P1, dis, b1, P0);
    k_agg_edges_t<16><<<nblk((long long)NEDGES * 16, T), T, 0, stream>>>(row, col, dis, P1, P0);
    k_zero<<<1, 128, 0, stream>>>(stat, 128);
    k_stats_t<16><<<nblk(NNODES, T), T, 0, stream>>>(P0, stat);
    k_coeff<<<1, 64, 0, stream>>>(stat, g1, be1, cf, 16);
    k_bnrelu<<<nblk((long long)NNODES * 16, T), T, 0, stream>>>(P0, cf, 16);

    // ---- layer 2: 16 -> 32
    k_gemm_wmma_t<16, 32><<<gemmBlocks, T, 0, stream>>>(P0, W2, P1, ntiles);
    k_agg_init_t<32><<<nblk((long long)NNODES * 32, T), T, 0, stream>>>(P1, dis, b2, P0);
    k_agg_edges_t<32><<<nblk((long long)NEDGES * 32, T), T, 0, stream>>>(row, col, dis, P1, P0);
    k_zero<<<1, 128, 0, stream>>>(stat, 128);
    k_stats_t<32><<<nblk(NNODES, T), T, 0, stream>>>(P0, stat);
    k_coeff<<<1, 64, 0, stream>>>(stat, g2, be2, cf, 32);
    k_bnrelu<<<nblk((long long)NNODES * 32, T), T, 0, stream>>>(P0, cf, 32);

    // ---- layer 3: 32 -> 64
    k_gemm_wmma_t<32, 64><<<gemmBlocks, T, 0, stream>>>(P0, W3, P1, ntiles);
    k_agg_init_t<64><<<nblk((long long)NNODES * 64, T), T, 0, stream>>>(P1, dis, b3, P0);
    k_agg_edges_t<64><<<nblk((long long)NEDGES * 64, T), T, 0, stream>>>(row, col, dis, P1, P0);
    k_zero<<<1, 128, 0, stream>>>(stat, 128);
    k_stats_t<64><<<nblk(NNODES, T), T, 0, stream>>>(P0, stat);
    k_coeff<<<1, 64, 0, stream>>>(stat, g3, be3, cf, 64);
    k_bnrelu<<<nblk((long long)NNODES * 64, T), T, 0, stream>>>(P0, cf, 64);

    // ---- global mean pool + FC
    k_zero<<<nblk(NB * 64 + NB, T), T, 0, stream>>>(pool, NB * 64 + NB); // pool then cnt (contiguous)
    k_pool<<<nblk((long long)NNODES * 64, T), T, 0, stream>>>(P0, bat, pool, cnt);
    k_fc<<<nblk(NB * 10, T), T, 0, stream>>>(pool, cnt, fcW, fcb, out);
}